// AgentModel_21517786153340
// MI455X (gfx1250) — compile-verified
//
#include <hip/hip_runtime.h>

// Problem constants (from reference): B=8, N=256, TD=LH=H=128, A=5, L=2
#define Bc   8
#define Nc   256
#define TDc  128
#define LHc  128
#define Hc   128
#define Ac   5
#define Lc   2
#define MROWS (Bc * Nc)          // 2048 total rows
#define KIN   (TDc + LHc)        // 256

typedef float v2f __attribute__((ext_vector_type(2)));
typedef float v8f __attribute__((ext_vector_type(8)));

// ---------------------------------------------------------------------------
// Concat theta (B,N,128) and cell (B,N,128) into x (B*N, 256)
// ---------------------------------------------------------------------------
__global__ __launch_bounds__(256)
void concat_kernel(const float* __restrict__ theta,
                   const float* __restrict__ cell,
                   float* __restrict__ x)
{
    int idx = blockIdx.x * blockDim.x + threadIdx.x;   // over MROWS*KIN
    int row = idx >> 8;
    int col = idx & 255;
    float v = (col < TDc) ? theta[row * TDc + col]
                          : cell[row * LHc + (col - TDc)];
    x[idx] = v;
}

// ---------------------------------------------------------------------------
// C[M x Nout] = act( A[M x K] * W[Nout x K]^T + bias )   (row-major everywhere)
// One wave per 16x16 output tile, V_WMMA_F32_16X16X4_F32 over the K loop.
//
// fp32 16x4 A layout: lanes 0-15 -> M=lane, VGPR0=K+0, VGPR1=K+1
//                     lanes 16-31 -> M=lane-16, VGPR0=K+2, VGPR1=K+3
// B (4x16) mirrors it with N striped across lanes; since B here is W^T,
// both A and W are read with the identical float2 pattern.
// C layout: VGPR v -> M = 16*tm + (lane>=16 ? 8 : 0) + v, N = 16*tn + (lane&15)
// ---------------------------------------------------------------------------
__global__ __launch_bounds__(256)
void gemm_wmma_f32(const float* __restrict__ A, int lda,
                   const float* __restrict__ W, int ldw,
                   const float* __restrict__ bias,
                   float* __restrict__ C, int ldc,
                   int K, int tiles_n, int do_relu)
{
    int wid  = (blockIdx.x * blockDim.x + threadIdx.x) >> 5;
    int lane = threadIdx.x & 31;
    int tm   = wid / tiles_n;
    int tn   = wid % tiles_n;
    int half = lane >> 4;       // 0: K+0/K+1 ; 1: K+2/K+3
    int l16  = lane & 15;

    const float* ap = A + (size_t)(tm * 16 + l16) * lda + half * 2;
    const float* wp = W + (size_t)(tn * 16 + l16) * ldw + half * 2;

    v8f acc = {0.f, 0.f, 0.f, 0.f, 0.f, 0.f, 0.f, 0.f};

#pragma unroll 4
    for (int k = 0; k < K; k += 4) {
        v2f av = *(const v2f*)(ap + k);
        v2f bv = *(const v2f*)(wp + k);
        // 8 args: (neg_a, A, neg_b, B, c_mod, C, reuse_a, reuse_b)
        acc = __builtin_amdgcn_wmma_f32_16x16x4_f32(
                  false, av, false, bv, (short)0, acc, false, false);
    }

    int ccol = tn * 16 + l16;
    int crow = tm * 16 + half * 8;
    float bv = bias ? bias[ccol] : 0.0f;

#pragma unroll
    for (int v = 0; v < 8; ++v) {
        float r = acc[v] + bv;
        if (do_relu) r = fmaxf(r, 0.0f);
        C[(size_t)(crow + v) * ldc + ccol] = r;
    }
}

// ---------------------------------------------------------------------------
// Message aggregation (relu is INSIDE the j-sum, so not expressible as GEMM):
//   agg[b,i,o] = (1/(N-1)) * ( sum_j relu(pj[b,j,o] + pi[b,i,o] + mb[o])
//                              - relu(pj[b,i,o] + pi[b,i,o] + mb[o]) )
//   hsum[b,i,o] = h[b,i,o] + agg[b,i,o]
//
// Block = (b, group of 16 i).  pj[b] is streamed through LDS in 64-row
// chunks (32 KB); each thread owns one channel o and 8 i-values, so every
// LDS read of pj is reused 8x in registers.
// ---------------------------------------------------------------------------
__global__ __launch_bounds__(256)
void agg_kernel(const float* __restrict__ pj,
                const float* __restrict__ pi,
                const float* __restrict__ h,
                const float* __restrict__ mb,
                float* __restrict__ hsum)
{
    __shared__ float lds[64 * Hc];          // 32 KB chunk of pj[b]

    int b  = blockIdx.x >> 4;               // N/16 = 16 i-groups per batch
    int ig = blockIdx.x & 15;

    int o     = threadIdx.x & (Hc - 1);
    int isub  = threadIdx.x >> 7;           // 0 or 1
    int i0    = ig * 16 + isub;             // this thread's i's: i0, i0+2, ..., i0+14

    const float* pjb = pj + (size_t)b * Nc * Hc;
    float mbo = mb[o];

    float cvals[8], svals[8];
#pragma unroll
    for (int t = 0; t < 8; ++t) {
        int i = i0 + 2 * t;
        cvals[t] = pi[((size_t)b * Nc + i) * Hc + o] + mbo;
        svals[t] = 0.0f;
    }

    for (int jc = 0; jc < Nc; jc += 64) {
        __syncthreads();
        const float4* src = (const float4*)(pjb + (size_t)jc * Hc);
        float4*       dst = (float4*)lds;
        for (int t = threadIdx.x; t < 64 * Hc / 4; t += 256) dst[t] = src[t];
        __syncthreads();

        for (int j = 0; j < 64; ++j) {
            float pv = lds[j * Hc + o];
#pragma unroll
            for (int t = 0; t < 8; ++t)
                svals[t] += fmaxf(pv + cvals[t], 0.0f);
        }
    }

#pragma unroll
    for (int t = 0; t < 8; ++t) {
        int i = i0 + 2 * t;
        size_t base = ((size_t)b * Nc + i) * Hc + o;
        float s = svals[t] - fmaxf(pjb[(size_t)i * Hc + o] + cvals[t], 0.0f);
        hsum[base] = h[base] + s * (1.0f / (float)(Nc - 1));
    }
}

// ---------------------------------------------------------------------------
// Final tiny projection: logits (2048 x 5) = z (2048 x 128) @ W2^T + b2
// ---------------------------------------------------------------------------
__global__ __launch_bounds__(256)
void head2_kernel(const float* __restrict__ z,
                  const float* __restrict__ W2,
                  const float* __restrict__ b2,
                  float* __restrict__ out)
{
    int idx = blockIdx.x * blockDim.x + threadIdx.x;
    if (idx >= MROWS * Ac) return;
    int row = idx / Ac;
    int a   = idx % Ac;
    const float* zr = z  + (size_t)row * Hc;
    const float* wr = W2 + (size_t)a * Hc;
    float s = b2[a];
#pragma unroll 8
    for (int k = 0; k < Hc; ++k) s += zr[k] * wr[k];
    out[idx] = s;
}

// ---------------------------------------------------------------------------
extern "C" void kernel_launch(void* const* d_in, const int* in_sizes, int n_in,
                              void* d_out, int out_size, void* d_ws, size_t ws_size,
                              hipStream_t stream)
{
    (void)in_sizes; (void)n_in; (void)out_size; (void)ws_size;

    const float* theta   = (const float*)d_in[0];
    const float* cell    = (const float*)d_in[1];
    const float* enc_W   = (const float*)d_in[2];
    const float* enc_b   = (const float*)d_in[3];
    const float* msg_W   = (const float*)d_in[4];   // (L, H, 2H)
    const float* msg_b   = (const float*)d_in[5];   // (L, H)
    const float* upd_W   = (const float*)d_in[6];   // (L, H, H)
    const float* upd_b   = (const float*)d_in[7];   // (L, H)
    const float* head_W1 = (const float*)d_in[8];
    const float* head_b1 = (const float*)d_in[9];
    const float* head_W2 = (const float*)d_in[10];
    const float* head_b2 = (const float*)d_in[11];
    float* out = (float*)d_out;

    // Workspace layout (floats)
    float* ws = (float*)d_ws;
    float* x    = ws;                               // 2048*256
    float* h    = x   + (size_t)MROWS * KIN;        // 2048*128
    float* pj   = h   + (size_t)MROWS * Hc;
    float* pi   = pj  + (size_t)MROWS * Hc;
    float* hsum = pi  + (size_t)MROWS * Hc;
    float* z    = hsum+ (size_t)MROWS * Hc;

    const int tiles_m = MROWS / 16;                 // 128
    const int tiles_n = Hc / 16;                    // 8
    const int gemm_blocks = tiles_m * tiles_n * 32 / 256;  // 128 blocks

    // 1) x = [theta | cell]
    concat_kernel<<<MROWS * KIN / 256, 256, 0, stream>>>(theta, cell, x);

    // 2) h = relu(x @ enc_W^T + enc_b)
    gemm_wmma_f32<<<gemm_blocks, 256, 0, stream>>>(
        x, KIN, enc_W, KIN, enc_b, h, Hc, KIN, tiles_n, 1);

    // 3) message-passing layers
    for (int l = 0; l < Lc; ++l) {
        const float* Wl  = msg_W + (size_t)l * Hc * 2 * Hc;
        const float* mbl = msg_b + (size_t)l * Hc;
        const float* Wu  = upd_W + (size_t)l * Hc * Hc;
        const float* bu  = upd_b + (size_t)l * Hc;

        // pj = h @ Wj^T  (Wj = msg_W[l][:, :H], row stride 2H)
        gemm_wmma_f32<<<gemm_blocks, 256, 0, stream>>>(
            h, Hc, Wl, 2 * Hc, nullptr, pj, Hc, Hc, tiles_n, 0);
        // pi = h @ Wi^T  (Wi = msg_W[l][:, H:], row stride 2H)
        gemm_wmma_f32<<<gemm_blocks, 256, 0, stream>>>(
            h, Hc, Wl + Hc, 2 * Hc, nullptr, pi, Hc, Hc, tiles_n, 0);

        // hsum = h + agg
        agg_kernel<<<Bc * (Nc / 16), 256, 0, stream>>>(pj, pi, h, mbl, hsum);

        // h = relu(hsum @ upd_W^T + upd_b)
        gemm_wmma_f32<<<gemm_blocks, 256, 0, stream>>>(
            hsum, Hc, Wu, Hc, bu, h, Hc, Hc, tiles_n, 1);
    }

    // 4) z = relu(h @ head_W1^T + head_b1)
    gemm_wmma_f32<<<gemm_blocks, 256, 0, stream>>>(
        h, Hc, head_W1, Hc, head_b1, z, Hc, Hc, tiles_n, 1);

    // 5) logits = z @ head_W2^T + head_b2
    head2_kernel<<<(MROWS * Ac + 255) / 256, 256, 0, stream>>>(
        z, head_W2, head_b2, out);
}